// SPDEncoder_22643067584697
// MI455X (gfx1250) — compile-verified
//
#include <hip/hip_runtime.h>
#include <math.h>

#define BATCH 2048
#define CH    64
#define TLEN  1000
#define DD    4
#define THR   1e-4f
#define EPS_BN 1e-5f

typedef __attribute__((ext_vector_type(2))) float v2f;
typedef __attribute__((ext_vector_type(8))) float v8f;

#if defined(__has_builtin)
#if __has_builtin(__builtin_amdgcn_wmma_f32_16x16x4_f32)
#define HAVE_WMMA_F32X4 1
#endif
#endif

// ---------------- 4x4 symmetric eig (cyclic Jacobi), per-thread ----------------
__device__ __forceinline__ void eigh4(const float* A, float w[4], float V[16]) {
    float M[16];
#pragma unroll
    for (int i = 0; i < 16; ++i) M[i] = A[i];
#pragma unroll
    for (int i = 0; i < 16; ++i) V[i] = (i % 5 == 0) ? 1.0f : 0.0f;
    for (int sweep = 0; sweep < 8; ++sweep) {
#pragma unroll
        for (int p = 0; p < 3; ++p) {
#pragma unroll
            for (int q = p + 1; q < 4; ++q) {
                float apq = M[p * 4 + q];
                if (fabsf(apq) < 1e-20f) continue;
                float app = M[p * 4 + p], aqq = M[q * 4 + q];
                float tau = (aqq - app) / (2.0f * apq);
                float t = ((tau >= 0.0f) ? 1.0f : -1.0f) /
                          (fabsf(tau) + sqrtf(1.0f + tau * tau));
                float c = 1.0f / sqrtf(1.0f + t * t);
                float s = t * c;
#pragma unroll
                for (int k = 0; k < 4; ++k) {   // M = M * J
                    float mkp = M[k * 4 + p], mkq = M[k * 4 + q];
                    M[k * 4 + p] = c * mkp - s * mkq;
                    M[k * 4 + q] = s * mkp + c * mkq;
                }
#pragma unroll
                for (int k = 0; k < 4; ++k) {   // M = J^T * M
                    float mpk = M[p * 4 + k], mqk = M[q * 4 + k];
                    M[p * 4 + k] = c * mpk - s * mqk;
                    M[q * 4 + k] = s * mpk + c * mqk;
                }
#pragma unroll
                for (int k = 0; k < 4; ++k) {   // V = V * J
                    float vkp = V[k * 4 + p], vkq = V[k * 4 + q];
                    V[k * 4 + p] = c * vkp - s * vkq;
                    V[k * 4 + q] = s * vkp + c * vkq;
                }
            }
        }
    }
#pragma unroll
    for (int i = 0; i < 4; ++i) w[i] = M[i * 4 + i];
}

// Out = V diag(f) V^T
__device__ __forceinline__ void recon4(const float V[16], const float f[4], float* O) {
#pragma unroll
    for (int i = 0; i < 4; ++i)
#pragma unroll
        for (int j = 0; j < 4; ++j) {
            float acc = 0.0f;
#pragma unroll
            for (int k = 0; k < 4; ++k) acc += V[i * 4 + k] * f[k] * V[j * 4 + k];
            O[i * 4 + j] = acc;
        }
}

__device__ __forceinline__ void mm4(const float* A, const float* B, float* C) {
#pragma unroll
    for (int i = 0; i < 4; ++i)
#pragma unroll
        for (int j = 0; j < 4; ++j) {
            float acc = 0.0f;
#pragma unroll
            for (int k = 0; k < 4; ++k) acc += A[i * 4 + k] * B[k * 4 + j];
            C[i * 4 + j] = acc;
        }
}

// ---------------- Phase A: streaming projection + 4x4 covariance ----------------
// spd[b] = Cov( W^T x[b] ) over time, using  (G - s s^T/T)/(T-1).
// One workgroup (8 waves) per batch element.
// Main loop: 62 full 16-timestep tiles, branch-free loads feeding a chain of 16
// V_WMMA_F32_16X16X4_F32 (K-blocks of 4 channels, 64 total). Tail (t=992..999)
// is done with scalar FMAs on threads 0-7 so WMMA never runs under partial EXEC.
__global__ __launch_bounds__(256) void covproj_kernel(const float* __restrict__ x,
                                                      const float* __restrict__ W,
                                                      float* __restrict__ spd) {
    const int b    = blockIdx.x;
    const int tid  = threadIdx.x;
    const int lane = tid & 31;
    const int wave = tid >> 5;
    const float* __restrict__ xb = x + (size_t)b * (CH * TLEN);

    float s0 = 0.f, s1 = 0.f, s2 = 0.f, s3 = 0.f;
    float g[10];
#pragma unroll
    for (int i = 0; i < 10; ++i) g[i] = 0.f;

#ifdef HAVE_WMMA_F32X4
    const int hi = lane >> 4;     // lane half (selects K offset per ISA layout)
    const int ln = lane & 15;     // M for A-frag, N (time) for B-frag
    const int k0 = hi ? 2 : 0;
    // A-fragments: A[m,k] = W^T[m, kb*4+k] = W[(kb*4+k)*4 + m], rows m>=4 are zero.
    // Layout (ISA 7.12.2, 32-bit A 16x4): VGPR0 -> K=0 (lanes0-15) / K=2 (lanes16-31);
    //                                     VGPR1 -> K=1 / K=3.
    v2f afrag[16];
#pragma unroll
    for (int kb = 0; kb < 16; ++kb) {
        float a0 = (ln < 4) ? W[(kb * 4 + k0) * 4 + ln] : 0.0f;
        float a1 = (ln < 4) ? W[(kb * 4 + k0 + 1) * 4 + ln] : 0.0f;
        v2f a; a.x = a0; a.y = a1;
        afrag[kb] = a;
    }

    // ---- 62 full tiles: t = tile*16 + ln <= 991 < 1000, no guards needed ----
    for (int tile = wave; tile < 62; tile += 8) {
        const float* __restrict__ p = xb + (tile * 16 + ln);
        float bx[16], by[16];
#pragma unroll
        for (int kb = 0; kb < 16; ++kb) {
            // B[k,n] = x[b, kb*4+k, t0+n]; VGPR0 -> K=0/2, VGPR1 -> K=1/3
            bx[kb] = __builtin_nontemporal_load(p + (size_t)(kb * 4 + k0) * TLEN);
            by[kb] = __builtin_nontemporal_load(p + (size_t)(kb * 4 + k0 + 1) * TLEN);
        }
        v8f d = {};                         // fresh accumulator per tile
#pragma unroll
        for (int kb = 0; kb < 16; ++kb) {
            v2f bf; bf.x = bx[kb]; bf.y = by[kb];
            d = __builtin_amdgcn_wmma_f32_16x16x4_f32(false, afrag[kb], false, bf,
                                                      (short)0, d, false, false);
        }
        // lanes 0-15: d[j] = y_j(t0+ln) for j=0..3; everything else is exactly 0.
        float y0 = d[0], y1 = d[1], y2 = d[2], y3 = d[3];
        s0 += y0; s1 += y1; s2 += y2; s3 += y3;
        g[0] += y0 * y0; g[1] += y0 * y1; g[2] += y0 * y2; g[3] += y0 * y3;
        g[4] += y1 * y1; g[5] += y1 * y2; g[6] += y1 * y3;
        g[7] += y2 * y2; g[8] += y2 * y3; g[9] += y3 * y3;
    }

    // ---- tail: t = 992..999, scalar dot products on threads 0..7 (no WMMA) ----
    if (tid < 8) {
        const int t = 992 + tid;
        float y0 = 0.f, y1 = 0.f, y2 = 0.f, y3 = 0.f;
#pragma unroll 8
        for (int c = 0; c < CH; ++c) {
            float xv = xb[(size_t)c * TLEN + t];
            y0 += xv * W[c * 4 + 0]; y1 += xv * W[c * 4 + 1];
            y2 += xv * W[c * 4 + 2]; y3 += xv * W[c * 4 + 3];
        }
        s0 += y0; s1 += y1; s2 += y2; s3 += y3;
        g[0] += y0 * y0; g[1] += y0 * y1; g[2] += y0 * y2; g[3] += y0 * y3;
        g[4] += y1 * y1; g[5] += y1 * y2; g[6] += y1 * y3;
        g[7] += y2 * y2; g[8] += y2 * y3; g[9] += y3 * y3;
    }
#else
    // Scalar fallback: per-thread timesteps
    for (int t = tid; t < TLEN; t += 256) {
        float y0 = 0.f, y1 = 0.f, y2 = 0.f, y3 = 0.f;
#pragma unroll 8
        for (int c = 0; c < CH; ++c) {
            float xv = __builtin_nontemporal_load(&xb[(size_t)c * TLEN + t]);
            y0 += xv * W[c * 4 + 0]; y1 += xv * W[c * 4 + 1];
            y2 += xv * W[c * 4 + 2]; y3 += xv * W[c * 4 + 3];
        }
        s0 += y0; s1 += y1; s2 += y2; s3 += y3;
        g[0] += y0 * y0; g[1] += y0 * y1; g[2] += y0 * y2; g[3] += y0 * y3;
        g[4] += y1 * y1; g[5] += y1 * y2; g[6] += y1 * y3;
        g[7] += y2 * y2; g[8] += y2 * y3; g[9] += y3 * y3;
    }
#endif

    // wave32 butterfly reduction of the 14 partials, then cross-wave via LDS
    float vals[14] = {s0, s1, s2, s3, g[0], g[1], g[2], g[3], g[4],
                      g[5], g[6], g[7], g[8], g[9]};
#pragma unroll
    for (int i = 0; i < 14; ++i)
        for (int off = 16; off > 0; off >>= 1)
            vals[i] += __shfl_xor(vals[i], off, 32);

    __shared__ float red[8][14];
    if (lane == 0)
#pragma unroll
        for (int i = 0; i < 14; ++i) red[wave][i] = vals[i];
    __syncthreads();

    if (tid == 0) {
        float tot[14];
#pragma unroll
        for (int i = 0; i < 14; ++i) {
            float a = 0.f;
            for (int wv = 0; wv < 8; ++wv) a += red[wv][i];
            tot[i] = a;
        }
        const float invT  = 1.0f / (float)TLEN;
        const float invT1 = 1.0f / (float)(TLEN - 1);
        float* o = spd + (size_t)b * 16;
        int tri = 4;
        for (int i = 0; i < 4; ++i)
            for (int j = i; j < 4; ++j) {
                float cv = (tot[tri++] - tot[i] * tot[j] * invT) * invT1;
                o[i * 4 + j] = cv;
                o[j * 4 + i] = cv;
            }
    }
}

// ---------------- Phase B1: ReEig (in-place) + partial sums of log(spd) ----------------
__global__ __launch_bounds__(256) void reeig_kernel(float* __restrict__ spd,
                                                    float* __restrict__ Lpart) {
    const int b = blockIdx.x * 256 + threadIdx.x;   // 8 * 256 = 2048
    float A[16];
#pragma unroll
    for (int i = 0; i < 16; ++i) A[i] = spd[(size_t)b * 16 + i];
    float w[4], V[16];
    eigh4(A, w, V);
    float wc[4], lw[4];
#pragma unroll
    for (int i = 0; i < 4; ++i) {
        wc[i] = fmaxf(w[i], THR);
        lw[i] = logf(wc[i]);
    }
    float Sc[16];
    recon4(V, wc, Sc);
#pragma unroll
    for (int i = 0; i < 16; ++i) spd[(size_t)b * 16 + i] = Sc[i];
    float LX[16];
    recon4(V, lw, LX);

    __shared__ float tmp[256];
    for (int cmp = 0; cmp < 16; ++cmp) {
        tmp[threadIdx.x] = LX[cmp];
        __syncthreads();
        for (int st = 128; st > 0; st >>= 1) {
            if (threadIdx.x < st) tmp[threadIdx.x] += tmp[threadIdx.x + st];
            __syncthreads();
        }
        if (threadIdx.x == 0) Lpart[blockIdx.x * 16 + cmp] = tmp[0];
        __syncthreads();
    }
}

// ---------------- Phase B2: G^{-1/2} from Lbar, and bias^{1/2} ----------------
__global__ void stats1_kernel(const float* __restrict__ Lpart,
                              const float* __restrict__ bn_bias,
                              float* __restrict__ Gis, float* __restrict__ Bs) {
    if (threadIdx.x != 0) return;
    float Lbar[16];
    for (int i = 0; i < 16; ++i) {
        float a = 0.f;
        for (int blk = 0; blk < 8; ++blk) a += Lpart[blk * 16 + i];
        Lbar[i] = a / (float)BATCH;
    }
    float w[4], V[16], f[4], O[16];
    eigh4(Lbar, w, V);
    for (int i = 0; i < 4; ++i) f[i] = expf(-0.5f * w[i]);
    recon4(V, f, O);
    for (int i = 0; i < 16; ++i) Gis[i] = O[i];

    float Bb[16];
    for (int i = 0; i < 16; ++i) Bb[i] = bn_bias[i];
    eigh4(Bb, w, V);
    for (int i = 0; i < 4; ++i) f[i] = sqrtf(fmaxf(w[i], 0.0f));
    recon4(V, f, O);
    for (int i = 0; i < 16; ++i) Bs[i] = O[i];
}

// ---------------- Phase B3: center at identity, eig, dispersion partials ----------------
__global__ __launch_bounds__(256) void center_kernel(const float* __restrict__ spd,
                                                     const float* __restrict__ Gis,
                                                     float* __restrict__ VXo,
                                                     float* __restrict__ wXo,
                                                     float* __restrict__ vpart) {
    const int b = blockIdx.x * 256 + threadIdx.x;
    float G[16], S[16], T1[16], Xc[16];
#pragma unroll
    for (int i = 0; i < 16; ++i) G[i] = Gis[i];
#pragma unroll
    for (int i = 0; i < 16; ++i) S[i] = spd[(size_t)b * 16 + i];
    mm4(G, S, T1);
    mm4(T1, G, Xc);
    // symmetrize for Jacobi robustness
#pragma unroll
    for (int i = 0; i < 4; ++i)
#pragma unroll
        for (int j = i + 1; j < 4; ++j) {
            float m = 0.5f * (Xc[i * 4 + j] + Xc[j * 4 + i]);
            Xc[i * 4 + j] = m; Xc[j * 4 + i] = m;
        }
    float w[4], V[16];
    eigh4(Xc, w, V);
#pragma unroll
    for (int i = 0; i < 16; ++i) VXo[(size_t)b * 16 + i] = V[i];
#pragma unroll
    for (int i = 0; i < 4; ++i) wXo[(size_t)b * 4 + i] = w[i];
    // ||log(Xc)||_F^2 = sum_i log(lambda_i)^2   (V orthogonal)
    float ssq = 0.f;
#pragma unroll
    for (int i = 0; i < 4; ++i) {
        float li = logf(fmaxf(w[i], 1e-30f));
        ssq += li * li;
    }
    __shared__ float tmp[256];
    tmp[threadIdx.x] = ssq;
    __syncthreads();
    for (int st = 128; st > 0; st >>= 1) {
        if (threadIdx.x < st) tmp[threadIdx.x] += tmp[threadIdx.x + st];
        __syncthreads();
    }
    if (threadIdx.x == 0) vpart[blockIdx.x] = tmp[0];
}

// ---------------- Phase B4: alpha = scale / sqrt(v + eps) ----------------
__global__ void stats2_kernel(const float* __restrict__ vpart,
                              const float* __restrict__ bn_scale,
                              float* __restrict__ alpha) {
    if (threadIdx.x != 0) return;
    float v = 0.f;
    for (int i = 0; i < 8; ++i) v += vpart[i];
    v /= (float)BATCH;
    alpha[0] = bn_scale[0] / sqrtf(v + EPS_BN);
}

// ---------------- Phase B5: Xn = exp(alpha*logXc) = V diag(l^alpha) V^T ; out ----------------
__global__ __launch_bounds__(256) void out_kernel(const float* __restrict__ VX,
                                                  const float* __restrict__ wX,
                                                  const float* __restrict__ Bs,
                                                  const float* __restrict__ alpha,
                                                  float* __restrict__ out) {
    const int b = blockIdx.x * 256 + threadIdx.x;
    float V[16], w[4];
#pragma unroll
    for (int i = 0; i < 16; ++i) V[i] = VX[(size_t)b * 16 + i];
#pragma unroll
    for (int i = 0; i < 4; ++i) w[i] = wX[(size_t)b * 4 + i];
    const float a = alpha[0];
    float f[4];
#pragma unroll
    for (int i = 0; i < 4; ++i) f[i] = expf(a * logf(fmaxf(w[i], 1e-30f)));
    float Xn[16];
    recon4(V, f, Xn);
    float Bm[16], T1[16], O[16];
#pragma unroll
    for (int i = 0; i < 16; ++i) Bm[i] = Bs[i];
    mm4(Bm, Xn, T1);
    mm4(T1, Bm, O);
#pragma unroll
    for (int i = 0; i < 4; ++i)
#pragma unroll
        for (int j = i + 1; j < 4; ++j) {
            float m = 0.5f * (O[i * 4 + j] + O[j * 4 + i]);
            O[i * 4 + j] = m; O[j * 4 + i] = m;
        }
    float wo[4], Vo[16];
    eigh4(O, wo, Vo);
    float lo[4];
#pragma unroll
    for (int i = 0; i < 4; ++i) lo[i] = logf(fmaxf(wo[i], 1e-30f));
    float Lm[16];
    recon4(Vo, lo, Lm);
    const float sq2 = 1.41421356237309515f;
    float* o = out + (size_t)b * 10;
    o[0] = Lm[0];        o[1] = Lm[1] * sq2;  o[2] = Lm[2] * sq2;  o[3] = Lm[3] * sq2;
    o[4] = Lm[5];        o[5] = Lm[6] * sq2;  o[6] = Lm[7] * sq2;
    o[7] = Lm[10];       o[8] = Lm[11] * sq2; o[9] = Lm[15];
}

extern "C" void kernel_launch(void* const* d_in, const int* in_sizes, int n_in,
                              void* d_out, int out_size, void* d_ws, size_t ws_size,
                              hipStream_t stream) {
    (void)in_sizes; (void)n_in; (void)out_size; (void)ws_size;
    const float* x        = (const float*)d_in[0];   // [2048, 64, 1000]
    const float* W        = (const float*)d_in[1];   // [64, 4]
    const float* bn_bias  = (const float*)d_in[2];   // [4, 4]
    const float* bn_scale = (const float*)d_in[3];   // scalar
    float* out = (float*)d_out;                      // [2048, 10]

    float* ws    = (float*)d_ws;
    float* spd   = ws;            // 2048*16
    float* VX    = ws + 32768;    // 2048*16
    float* wX    = ws + 65536;    // 2048*4
    float* Lpart = ws + 73728;    // 8*16
    float* vpart = ws + 73856;    // 8
    float* Gis   = ws + 73864;    // 16
    float* Bsb   = ws + 73880;    // 16
    float* alpha = ws + 73896;    // 1

    covproj_kernel<<<BATCH, 256, 0, stream>>>(x, W, spd);
    reeig_kernel <<<BATCH / 256, 256, 0, stream>>>(spd, Lpart);
    stats1_kernel<<<1, 32, 0, stream>>>(Lpart, bn_bias, Gis, Bsb);
    center_kernel<<<BATCH / 256, 256, 0, stream>>>(spd, Gis, VX, wX, vpart);
    stats2_kernel<<<1, 32, 0, stream>>>(vpart, bn_scale, alpha);
    out_kernel   <<<BATCH / 256, 256, 0, stream>>>(VX, wX, Bsb, alpha, out);
}